// attention_shuffle_4741643895143
// MI455X (gfx1250) — compile-verified
//
#include <hip/hip_runtime.h>
#include <math.h>

// ---------------------------------------------------------------------------
// Problem: B=32, C=512, HW=4096 (64x64), f32.
//   pooled[b,c] = mean_hw x[b,c,:,:]
//   y = conv1d_3tap(pooled, w, pad=1); wgt = sigmoid(y)
//   idx = argsort(-wgt) (stable, per batch)
//   out[b,i,:,:] = x[b,idx[b,i],:,:]
// Memory-bound: ~768MB traffic -> ~33us at 23.3 TB/s.
// ---------------------------------------------------------------------------

#define ECA_B  32
#define ECA_C  512
#define ECA_HW 4096

typedef __attribute__((ext_vector_type(2))) float v2f;
typedef __attribute__((ext_vector_type(8))) float v8f;
typedef __attribute__((ext_vector_type(4))) int   v4i;

#if __has_builtin(__builtin_amdgcn_wmma_f32_16x16x4_f32)
#define ECA_USE_WMMA 1
#endif

#if __has_builtin(__builtin_amdgcn_global_load_async_to_lds_b128) && \
    __has_builtin(__builtin_amdgcn_global_store_async_from_lds_b128) && \
    __has_builtin(__builtin_amdgcn_s_wait_asynccnt)
#define ECA_USE_ASYNC_LDS 1
#endif

// ---------------------------------------------------------------------------
// Kernel 1: channel mean pooling.
// One wave (32 lanes) handles 16 channels. Each lane f32-accumulates two
// 1024-element chunks of its channel (its two A-matrix k-slots), then a single
// V_WMMA_F32_16X16X4_F32 with B=ones reduces across k: D[m][n] = sum_k A[m][k]
// = full 4096-element channel sum (same value in every column n).
// A layout (ISA 7.12.2, 32-bit A 16x4): lane L<16 -> A[L][0..1], lane L>=16 ->
// A[L-16][2..3]. D layout: lane 0 holds rows M=0..7 in v[0..7] (N=0), lane 16
// holds rows M=8..15.
// Grid: 1024 waves = 128 blocks x 256 threads.
// ---------------------------------------------------------------------------
__global__ void eca_pool_kernel(const float* __restrict__ x,
                                float* __restrict__ pooled) {
    const int wave = (blockIdx.x * blockDim.x + threadIdx.x) >> 5; // group id
    const int lane = threadIdx.x & 31;
    const int m    = lane & 15;      // row (channel within group)
    const int kh   = lane >> 4;      // 0 -> k-slots {0,1}, 1 -> {2,3}

    const size_t chFlat = (size_t)wave * 16 + m;          // flat b*512+c
    const float4* p0 = (const float4*)(x + chFlat * ECA_HW + (size_t)(2 * kh) * 1024);
    const float4* p1 = (const float4*)(x + chFlat * ECA_HW + (size_t)(2 * kh + 1) * 1024);

    float s0a = 0.f, s0b = 0.f, s1a = 0.f, s1b = 0.f;
#pragma unroll 4
    for (int i = 0; i < 256; i += 2) {
        float4 u0 = p0[i];
        float4 u1 = p0[i + 1];
        float4 v0 = p1[i];
        float4 v1 = p1[i + 1];
        s0a += (u0.x + u0.y) + (u0.z + u0.w);
        s0b += (u1.x + u1.y) + (u1.z + u1.w);
        s1a += (v0.x + v0.y) + (v0.z + v0.w);
        s1b += (v1.x + v1.y) + (v1.z + v1.w);
    }
    const float a0 = s0a + s0b;   // partial for k-slot 2*kh
    const float a1 = s1a + s1b;   // partial for k-slot 2*kh+1

#if defined(ECA_USE_WMMA) && defined(__AMDGCN__)
    v2f a = {a0, a1};
    v2f bones = {1.0f, 1.0f};     // B = all-ones 4x16 -> row-sum broadcast
    v8f c = {};
    v8f d = __builtin_amdgcn_wmma_f32_16x16x4_f32(
        /*neg_a=*/false, a, /*neg_b=*/false, bones,
        /*c_mod=*/(short)0, c, /*reuse_a=*/false, /*reuse_b=*/false);
    if (lane == 0 || lane == 16) {
        const size_t base = (size_t)wave * 16 + (kh ? 8 : 0);
#pragma unroll
        for (int e = 0; e < 8; ++e)
            pooled[base + e] = d[e] * (1.0f / (float)ECA_HW);
    }
#else
    // Fallback: cross-half-wave shuffle reduction.
    float s = a0 + a1;
    s += __shfl_down(s, 16, 32);
    if (lane < 16)
        pooled[(size_t)wave * 16 + m] = s * (1.0f / (float)ECA_HW);
#endif
}

// ---------------------------------------------------------------------------
// Kernel 2: per-batch conv1d(3) + sigmoid + stable descending argsort.
// One block per batch, 256 threads, bitonic sort of 512 packed 64-bit keys:
//   key = (~orderedBits(sigmoid(y)) << 32) | channel
// Ascending sort of key == descending weight, ties -> lower channel first
// (matches stable jnp.argsort of -weight).
// ---------------------------------------------------------------------------
__global__ void eca_sort_kernel(const float* __restrict__ pooled,
                                const float* __restrict__ w,
                                int* __restrict__ idxOut) {
    __shared__ float pbuf[ECA_C];
    __shared__ unsigned long long keys[ECA_C];

    const int b   = blockIdx.x;
    const int tid = threadIdx.x;
    const float w0 = w[0], w1 = w[1], w2 = w[2];

    for (int c = tid; c < ECA_C; c += 256)
        pbuf[c] = pooled[(size_t)b * ECA_C + c];
    __syncthreads();

    for (int c = tid; c < ECA_C; c += 256) {
        const float left  = (c > 0)         ? pbuf[c - 1] : 0.0f;
        const float mid   = pbuf[c];
        const float right = (c < ECA_C - 1) ? pbuf[c + 1] : 0.0f;
        const float y   = w0 * left + w1 * mid + w2 * right;
        const float wgt = 1.0f / (1.0f + expf(-y));
        unsigned u  = __float_as_uint(wgt);
        unsigned ou = (u & 0x80000000u) ? ~u : (u | 0x80000000u); // ascending order map
        unsigned dk = ~ou;                                        // descending order map
        keys[c] = ((unsigned long long)dk << 32) | (unsigned)c;
    }

    // Bitonic sort, 512 elements, ascending.
    for (int k = 2; k <= ECA_C; k <<= 1) {
        for (int j = k >> 1; j > 0; j >>= 1) {
            __syncthreads();
            for (int t = tid; t < ECA_C; t += 256) {
                const int ixj = t ^ j;
                if (ixj > t) {
                    const unsigned long long va = keys[t];
                    const unsigned long long vb = keys[ixj];
                    const bool asc = ((t & k) == 0);
                    if ((va > vb) == asc) { keys[t] = vb; keys[ixj] = va; }
                }
            }
        }
    }
    __syncthreads();

    for (int i = tid; i < ECA_C; i += 256)
        idxOut[(size_t)b * ECA_C + i] = (int)(keys[i] & 0xFFFFFFFFull);
}

// ---------------------------------------------------------------------------
// Kernel 3: channel permute (gather). One block per output channel row
// (16 KB). Preferred path: gfx1250 async LDS copy (VGPR-free DMA through LDS,
// tracked by ASYNCcnt); fallback: plain b128 load/store streaming copy.
// ---------------------------------------------------------------------------
__global__ void eca_gather_kernel(const float* __restrict__ x,
                                  const int* __restrict__ idx,
                                  float* __restrict__ out) {
    const int row = blockIdx.x;            // flat b*512 + i
    const int b   = row >> 9;
    const int src = idx[row];              // uniform across block
    const size_t srcOff = ((size_t)b * ECA_C + (size_t)src) * ECA_HW;
    const size_t dstOff = (size_t)row * ECA_HW;

    const float4* s = (const float4*)(x + srcOff);
    float4*       d = (float4*)(out + dstOff);
    const int tid = threadIdx.x;

#if defined(ECA_USE_ASYNC_LDS) && defined(__AMDGCN__)
    __shared__ float4 tile[1024];          // 16 KB staging
#pragma unroll
    for (int r = 0; r < 4; ++r) {
        __builtin_amdgcn_global_load_async_to_lds_b128(
            (__attribute__((address_space(1))) v4i*)(s + tid + 256 * r),
            (__attribute__((address_space(3))) v4i*)&tile[tid + 256 * r],
            0, 0);
    }
    __builtin_amdgcn_s_wait_asynccnt(0);   // LDS writes visible to our stores
#pragma unroll
    for (int r = 0; r < 4; ++r) {
        __builtin_amdgcn_global_store_async_from_lds_b128(
            (__attribute__((address_space(1))) v4i*)(d + tid + 256 * r),
            (__attribute__((address_space(3))) v4i*)&tile[tid + 256 * r],
            0, 0);
    }
    __builtin_amdgcn_s_wait_asynccnt(0);
#else
#pragma unroll
    for (int r = 0; r < 4; ++r)
        d[tid + 256 * r] = s[tid + 256 * r];
#endif
}

// ---------------------------------------------------------------------------
// Host launcher.
// d_in[0]: x  (32*512*64*64 f32)   d_in[1]: w (3 f32)
// d_out:   out (32*512*64*64 f32)
// d_ws:    [0,64KB) pooled f32[16384]; [64KB,128KB) indices i32[16384]
// ---------------------------------------------------------------------------
extern "C" void kernel_launch(void* const* d_in, const int* in_sizes, int n_in,
                              void* d_out, int out_size, void* d_ws, size_t ws_size,
                              hipStream_t stream) {
    (void)in_sizes; (void)n_in; (void)out_size; (void)ws_size;
    const float* x = (const float*)d_in[0];
    const float* w = (const float*)d_in[1];
    float* out = (float*)d_out;

    float* pooled = (float*)d_ws;
    int*   idx    = (int*)((char*)d_ws + (size_t)ECA_B * ECA_C * sizeof(float));

    // 1) pooling: 1024 waves = 128 blocks x 256 threads
    eca_pool_kernel<<<dim3((ECA_B * ECA_C / 16) / 8), dim3(256), 0, stream>>>(x, pooled);
    // 2) conv + sigmoid + argsort: one block per batch
    eca_sort_kernel<<<dim3(ECA_B), dim3(256), 0, stream>>>(pooled, w, idx);
    // 3) channel permute: one block per 16KB row
    eca_gather_kernel<<<dim3(ECA_B * ECA_C), dim3(256), 0, stream>>>(x, idx, out);
}